// GCN_86947317940591
// MI455X (gfx1250) — compile-verified
//
#include <hip/hip_runtime.h>
#include <hip/hip_bf16.h>
#include <math.h>

#define N_NODES   100000
#define N_EDGES   1600000
#define N_GRAPHS  64
#define F_IN      86
#define HDIM      128
#define N_CLASSES 18

typedef float v2f __attribute__((ext_vector_type(2)));
typedef float v8f __attribute__((ext_vector_type(8)));

static __device__ __forceinline__ void atomic_add_f32(float* p, float v) {
  __hip_atomic_fetch_add(p, v, __ATOMIC_RELAXED, __HIP_MEMORY_SCOPE_AGENT);
}

// ---------------- degree / dinv ----------------
__global__ void k_fill(float* p, float v, int n) {
  int i = blockIdx.x * blockDim.x + threadIdx.x;
  if (i < n) p[i] = v;
}

__global__ void k_deg(const int* __restrict__ dst, const float* __restrict__ ew,
                      float* __restrict__ deg, int e) {
  int i = blockIdx.x * blockDim.x + threadIdx.x;
  if (i < e) atomic_add_f32(&deg[dst[i]], ew[i]);
}

__global__ void k_rsqrt_inplace(float* d, int n) {
  int i = blockIdx.x * blockDim.x + threadIdx.x;
  if (i < n) d[i] = rsqrtf(d[i]);
}

// ------------- WMMA GEMM + fused self-loop epilogue --------------------------
// P[M,128]    = act(A[M,K]) @ W[K,128]
// out[M,128]  = dinv[m]^2 * P[m, :] + bias[:]        (GCN self-loop term)
//
// blockDim = 128 (4 waves). Each wave computes a 16x128 strip via 8 WMMA
// 16x16 tiles with V_WMMA_F32_16X16X4_F32 (exact fp32).  K and ACT are
// compile-time so the A-fragment array is fully SROA'd into VGPRs (no scratch
// spills).  W staged in LDS with row stride 136 floats (136 % 64 == 8 -> the
// two lane halves of a wave hit disjoint bank sets).
#define LW 136
template <int K, int ACT>
__global__ __launch_bounds__(128) void k_matmul(const float* __restrict__ A,
                                                const float* __restrict__ W,
                                                float* __restrict__ P,
                                                const float* __restrict__ dinv,
                                                const float* __restrict__ bias,
                                                float* __restrict__ out,
                                                int M) {
  constexpr int Kpad = (K + 3) & ~3;
  constexpr int NK   = Kpad >> 2;

  extern __shared__ float ldsW[];
  const int tid = threadIdx.x;

  // stage weights into LDS, zero-pad rows K..Kpad-1
  for (int idx = tid; idx < Kpad * HDIM; idx += 128) {
    int r = idx >> 7, c = idx & 127;
    ldsW[r * LW + c] = (r < K) ? W[r * HDIM + c] : 0.0f;
  }
  __syncthreads();

  const int wave = tid >> 5, lane = tid & 31;
  const int row0 = blockIdx.x * 64 + wave * 16;
  const int m    = lane & 15;       // M index within tile
  const int hh   = lane >> 4;       // lane-half selects K pair {0,1} vs {2,3}
  const int row  = row0 + m;
  const bool rv  = row < M;
  const float* arow = A + (size_t)row * K;

  // preload all A fragments for this wave's 16-row strip (<= 64 VGPRs)
  v2f afrag[NK];
#pragma unroll
  for (int i = 0; i < NK; ++i) {
    const int c = 4 * i + 2 * hh;
    float x0 = 0.0f, x1 = 0.0f;
    if (rv && c     < K) x0 = arow[c];
    if (rv && c + 1 < K) x1 = arow[c + 1];
    if (ACT) { x0 = fmaxf(x0, 0.0f); x1 = fmaxf(x1, 0.0f); }
    afrag[i].x = x0;
    afrag[i].y = x1;
  }

  const int colb  = lane & 15;
  const int kk    = hh << 1;
  const int rbase = row0 + (hh << 3);

  for (int t = 0; t < 8; ++t) {           // 8 N-tiles of 16 cols
    const int n0 = t << 4;
    v8f acc = {0.f, 0.f, 0.f, 0.f, 0.f, 0.f, 0.f, 0.f};
#pragma unroll
    for (int i = 0; i < NK; ++i) {
      const int k0 = (i << 2) + kk;
      v2f b;
      b.x = ldsW[(k0    ) * LW + n0 + colb];
      b.y = ldsW[(k0 + 1) * LW + n0 + colb];
      acc = __builtin_amdgcn_wmma_f32_16x16x4_f32(
          false, afrag[i], false, b, (short)0, acc, false, false);
    }
    // D layout: vgpr r -> M=r (lanes 0-15) / M=r+8 (lanes 16-31), N = n0+lane%16
    const int n  = n0 + colb;
    const float bn = bias[n];
#pragma unroll
    for (int r = 0; r < 8; ++r) {
      const int rr = rbase + r;
      if (rr < M) {
        const float v  = acc[r];
        const float di = dinv[rr];
        P[(size_t)rr * HDIM + n]   = v;
        out[(size_t)rr * HDIM + n] = di * di * v + bn;   // self-loop + bias
      }
    }
  }
}

// ---------------- edge scatter: out[dst] += norm * P[src] (wave per edge) ----
__global__ void k_scatter(const int* __restrict__ srcI, const int* __restrict__ dstI,
                          const float* __restrict__ ew, const float* __restrict__ dinv,
                          const float* __restrict__ P, float* __restrict__ out, int e) {
  int wid  = (blockIdx.x * blockDim.x + threadIdx.x) >> 5;
  int lane = threadIdx.x & 31;
  if (wid >= e) return;
  int s = srcI[wid], d = dstI[wid];
  float norm = dinv[s] * ew[wid] * dinv[d];
  const float* ps = P + (size_t)s * HDIM;
  float* po = out + (size_t)d * HDIM;
#pragma unroll
  for (int j = 0; j < 4; ++j) {
    int f = lane + 32 * j;
    atomic_add_f32(&po[f], norm * ps[f]);
  }
}

// ---------------- global add pool ----------------
__global__ void k_pool(const float* __restrict__ h, const int* __restrict__ batch,
                       float* __restrict__ g, int n) {
  int idx = blockIdx.x * blockDim.x + threadIdx.x;
  if (idx < n * HDIM) {
    int node = idx >> 7, f = idx & 127;
    atomic_add_f32(&g[batch[node] * HDIM + f], h[idx]);
  }
}

// ---------------- head: BN -> fc1 -> relu -> fc2 -> log_softmax ----------------
__global__ __launch_bounds__(128) void k_head(
    const float* __restrict__ g,
    const float* __restrict__ gamma, const float* __restrict__ beta,
    const float* __restrict__ mean,  const float* __restrict__ var,
    const float* __restrict__ fc1w,  const float* __restrict__ fc1b,
    const float* __restrict__ fc2w,  const float* __restrict__ fc2b,
    float* __restrict__ out) {
  __shared__ float v[HDIM];
  __shared__ float y1[HDIM];
  __shared__ float logits[N_CLASSES];
  __shared__ float red[2];
  const int gi = blockIdx.x, t = threadIdx.x;

  v[t] = (g[gi * HDIM + t] - mean[t]) * rsqrtf(var[t] + 1e-5f) * gamma[t] + beta[t];
  __syncthreads();

  float acc = fc1b[t];
  for (int f = 0; f < HDIM; ++f) acc = fmaf(v[f], fc1w[f * HDIM + t], acc);
  y1[t] = fmaxf(acc, 0.0f);
  __syncthreads();

  if (t < N_CLASSES) {
    float a = fc2b[t];
    for (int j = 0; j < HDIM; ++j) a = fmaf(y1[j], fc2w[j * N_CLASSES + t], a);
    logits[t] = a;
  }
  __syncthreads();

  if (t == 0) {
    float mx = logits[0];
    for (int c = 1; c < N_CLASSES; ++c) mx = fmaxf(mx, logits[c]);
    float s = 0.0f;
    for (int c = 0; c < N_CLASSES; ++c) s += __expf(logits[c] - mx);
    red[0] = mx;
    red[1] = __logf(s);
  }
  __syncthreads();

  if (t < N_CLASSES) out[gi * N_CLASSES + t] = logits[t] - red[0] - red[1];
}

// ---------------- launcher ----------------
extern "C" void kernel_launch(void* const* d_in, const int* in_sizes, int n_in,
                              void* d_out, int out_size, void* d_ws, size_t ws_size,
                              hipStream_t stream) {
  const float* x     = (const float*)d_in[0];
  const int*   ei    = (const int*)  d_in[1];
  const float* ew    = (const float*)d_in[2];
  const int*   batch = (const int*)  d_in[3];
  const float* W0    = (const float*)d_in[4];
  const float* b0    = (const float*)d_in[5];
  const float* Wh    = (const float*)d_in[6];
  const float* bh    = (const float*)d_in[7];
  const float* Wl    = (const float*)d_in[8];
  const float* bl    = (const float*)d_in[9];
  const float* fc1w  = (const float*)d_in[10];
  const float* fc1b  = (const float*)d_in[11];
  const float* fc2w  = (const float*)d_in[12];
  const float* fc2b  = (const float*)d_in[13];
  const float* gamma = (const float*)d_in[14];
  const float* beta  = (const float*)d_in[15];
  const float* mean  = (const float*)d_in[16];
  const float* var   = (const float*)d_in[17];
  float* out = (float*)d_out;

  // workspace layout (~103 MB of floats)
  float* P    = (float*)d_ws;                       // [N_NODES, 128] matmul result
  float* buf  = P    + (size_t)N_NODES * HDIM;      // [N_NODES, 128] layer output
  float* dinv = buf  + (size_t)N_NODES * HDIM;      // [N_NODES]
  float* g    = dinv + N_NODES;                     // [64, 128]

  const int* srcI = ei;
  const int* dstI = ei + N_EDGES;

  const int TB = 256;
  const int nodeGrid = (N_NODES + TB - 1) / TB;
  const int edgeGrid = (N_EDGES + TB - 1) / TB;
  const int featGrid = ((size_t)N_NODES * HDIM + TB - 1) / TB;
  const int mmGrid   = (N_NODES + 63) / 64;

  const size_t sh86  = (size_t)((F_IN + 3) & ~3) * LW * sizeof(float);
  const size_t sh128 = (size_t)HDIM * LW * sizeof(float);

  // degrees: dinv = rsqrt(1 + sum_e ew[dst==n])
  k_fill<<<nodeGrid, TB, 0, stream>>>(dinv, 1.0f, N_NODES);
  k_deg<<<edgeGrid, TB, 0, stream>>>(dstI, ew, dinv, N_EDGES);
  k_rsqrt_inplace<<<nodeGrid, TB, 0, stream>>>(dinv, N_NODES);

  for (int l = 0; l < 6; ++l) {
    if (l == 0) {
      k_matmul<F_IN, 0><<<mmGrid, 128, sh86, stream>>>(x, W0, P, dinv, b0, buf, N_NODES);
    } else if (l <= 4) {
      const float* W = Wh + (size_t)(l - 1) * HDIM * HDIM;
      const float* b = bh + (size_t)(l - 1) * HDIM;
      k_matmul<HDIM, 1><<<mmGrid, 128, sh128, stream>>>(buf, W, P, dinv, b, buf, N_NODES);
    } else {
      k_matmul<HDIM, 1><<<mmGrid, 128, sh128, stream>>>(buf, Wl, P, dinv, bl, buf, N_NODES);
    }
    k_scatter<<<(N_EDGES + 7) / 8, TB, 0, stream>>>(srcI, dstI, ew, dinv, P, buf, N_EDGES);
  }

  // pool + head
  k_fill<<<(N_GRAPHS * HDIM + TB - 1) / TB, TB, 0, stream>>>(g, 0.0f, N_GRAPHS * HDIM);
  k_pool<<<featGrid, TB, 0, stream>>>(buf, batch, g, N_NODES);
  k_head<<<N_GRAPHS, HDIM, 0, stream>>>(g, gamma, beta, mean, var,
                                        fc1w, fc1b, fc2w, fc2b, out);
}